// BatchedDistance_86517821213754
// MI455X (gfx1250) — compile-verified
//
#include <hip/hip_runtime.h>
#include <math.h>
#include <stdint.h>

typedef __attribute__((ext_vector_type(2))) float v2f;
typedef __attribute__((ext_vector_type(8))) float v8f;
typedef __attribute__((ext_vector_type(4))) int   v4i;

typedef __attribute__((address_space(1))) v4i* gas_v4i_ptr;   // global int4*
typedef __attribute__((address_space(3))) v4i* las_v4i_ptr;   // LDS int4*

#define NQ  8192
#define NR  32768
#define DIM 128

// Block tile: 128 (M) x 64 (N); 8 waves as 4x2 grid, each wave 32x32.
// Full K=128 staged in LDS once (~101 KB, fine under CDNA5's 320 KB/WGP).
#define BM 128
#define BN 64
#define LDSS 132       // 128 + 4 pad -> stride mod 64 banks = 4 (conflict-free frags)

__global__ __launch_bounds__(256)
void norms_kernel(const float* __restrict__ q, const float* __restrict__ r,
                  float* __restrict__ ws) {
    int i = blockIdx.x * blockDim.x + threadIdx.x;
    if (i >= NQ + NR) return;
    const float* src = (i < NQ) ? (q + (size_t)i * DIM) : (r + (size_t)(i - NQ) * DIM);
    const float4* p = (const float4*)src;
    float acc = 0.f;
#pragma unroll
    for (int j = 0; j < DIM / 4; ++j) {
        float4 v = p[j];
        acc += v.x * v.x + v.y * v.y + v.z * v.z + v.w * v.w;
    }
    ws[i] = acc;
}

// Async global -> LDS copy of 16 bytes per lane (CDNA5 GLOBAL_LOAD_ASYNC_TO_LDS_B128).
__device__ __forceinline__ void async_copy16(const float* g, float* l) {
    __builtin_amdgcn_global_load_async_to_lds_b128(
        (gas_v4i_ptr)(uintptr_t)g,
        (las_v4i_ptr)(uintptr_t)l,
        0, 0);
}

__global__ __launch_bounds__(256)
void dist_kernel(const float* __restrict__ Q, const float* __restrict__ R,
                 const float* __restrict__ qsq, const float* __restrict__ rsq,
                 float* __restrict__ out) {
    __shared__ float As[BM * LDSS];   // 67,584 B
    __shared__ float Bs[BN * LDSS];   // 33,792 B  (total ~101 KB)

    const int tid  = threadIdx.x;
    const int lane = tid & 31;
    const int wid  = tid >> 5;                  // 0..7
    const int m0   = blockIdx.y * BM;
    const int n0   = blockIdx.x * BN;

    const int wm   = (wid >> 1) * 32;           // wave row offset in tile
    const int wn   = (wid & 1) * 32;            // wave col offset in tile

    const int half = lane >> 4;                 // 0|1
    const int l15  = lane & 15;
    const int koff = half * 2;                  // K pair selector for A/B frags

    // ---- stage full-K A (128x128) and B (64x128) tiles via async b128 ----
#pragma unroll
    for (int i = 0; i < (BM * (DIM / 4)) / 256; ++i) {        // 16 iters
        int idx = tid + i * 256;
        int row = idx >> 5;                                    // DIM/4 = 32 float4 per row
        int c4  = idx & 31;
        async_copy16(Q + (size_t)(m0 + row) * DIM + c4 * 4,
                     As + row * LDSS + c4 * 4);
    }
#pragma unroll
    for (int i = 0; i < (BN * (DIM / 4)) / 256; ++i) {        // 8 iters
        int idx = tid + i * 256;
        int row = idx >> 5;
        int c4  = idx & 31;
        async_copy16(R + (size_t)(n0 + row) * DIM + c4 * 4,
                     Bs + row * LDSS + c4 * 4);
    }
    asm volatile("s_wait_asynccnt 0" ::: "memory");
    __syncthreads();

    // ---- K loop: 32 x (4 WMMA) out of LDS ----
    v8f c00 = {}, c01 = {}, c10 = {}, c11 = {};

    const float* Arow0 = As + (wm +      l15) * LDSS + koff;
    const float* Arow1 = As + (wm + 16 + l15) * LDSS + koff;
    const float* Brow0 = Bs + (wn +      l15) * LDSS + koff;
    const float* Brow1 = Bs + (wn + 16 + l15) * LDSS + koff;

#pragma unroll 4
    for (int k = 0; k < DIM; k += 4) {
        v2f a0 = *(const v2f*)(Arow0 + k);
        v2f a1 = *(const v2f*)(Arow1 + k);
        v2f b0 = *(const v2f*)(Brow0 + k);
        v2f b1 = *(const v2f*)(Brow1 + k);
        c00 = __builtin_amdgcn_wmma_f32_16x16x4_f32(false, a0, false, b0, (short)0, c00, false, false);
        c01 = __builtin_amdgcn_wmma_f32_16x16x4_f32(false, a0, false, b1, (short)0, c01, false, false);
        c10 = __builtin_amdgcn_wmma_f32_16x16x4_f32(false, a1, false, b0, (short)0, c10, false, false);
        c11 = __builtin_amdgcn_wmma_f32_16x16x4_f32(false, a1, false, b1, (short)0, c11, false, false);
    }

    // ---- epilogue: dist = sqrt(max(q^2 + r^2 - 2*cross, 0)) ----
    const int  ncol0 = n0 + wn + l15;
    const int  ncol1 = n0 + wn + 16 + l15;
    const float rs0  = rsq[ncol0];
    const float rs1  = rsq[ncol1];
    const int  mb0   = m0 + wm + half * 8;        // C layout: VGPR v -> M=v (+8 upper half)
    const int  mb1   = m0 + wm + 16 + half * 8;

#pragma unroll
    for (int v = 0; v < 8; ++v) {
        const int mrow0 = mb0 + v;
        const int mrow1 = mb1 + v;
        const float qs0 = qsq[mrow0];
        const float qs1 = qsq[mrow1];
        float d;
        d = qs0 + rs0 - 2.f * c00[v];
        out[(size_t)mrow0 * NR + ncol0] = sqrtf(fmaxf(d, 0.f));
        d = qs0 + rs1 - 2.f * c01[v];
        out[(size_t)mrow0 * NR + ncol1] = sqrtf(fmaxf(d, 0.f));
        d = qs1 + rs0 - 2.f * c10[v];
        out[(size_t)mrow1 * NR + ncol0] = sqrtf(fmaxf(d, 0.f));
        d = qs1 + rs1 - 2.f * c11[v];
        out[(size_t)mrow1 * NR + ncol1] = sqrtf(fmaxf(d, 0.f));
    }
}

extern "C" void kernel_launch(void* const* d_in, const int* in_sizes, int n_in,
                              void* d_out, int out_size, void* d_ws, size_t ws_size,
                              hipStream_t stream) {
    const float* q  = (const float*)d_in[0];
    const float* r  = (const float*)d_in[1];
    float* ws  = (float*)d_ws;          // [0..NQ) = q_sq, [NQ..NQ+NR) = r_sq
    float* out = (float*)d_out;

    norms_kernel<<<(NQ + NR + 255) / 256, 256, 0, stream>>>(q, r, ws);

    dim3 grid(NR / BN, NQ / BM);        // (512, 64)
    dist_kernel<<<grid, dim3(256), 0, stream>>>(q, r, ws, ws + NQ, out);
}